// local_rel_trafo_25305947308337
// MI455X (gfx1250) — compile-verified
//
#include <hip/hip_runtime.h>

typedef __attribute__((ext_vector_type(2))) float v2f;
typedef __attribute__((ext_vector_type(8))) float v8f;

// Blade bitmask per GATr index i (sorted by grade, then bitmask), packed as nibbles.
// BLADES = {0,1,2,4,8, 3,5,6,9,10,12, 7,11,13,14, 15}
#define BLADE_PACK 0xFEDB7CA965384210ULL
// Inverse map: index of blade bitmask, packed as nibbles.
// IDX[bm 0..15] = {0,1,2,5,3,6,7,11,4,8,9,12,10,13,14,15}
#define IDX_PACK   0xFEDAC984B7635210ULL

__device__ __forceinline__ unsigned nib(unsigned long long p, unsigned i) {
    return (unsigned)((p >> (4u * i)) & 0xFull);
}

// M_n[i][k] = GP[i, j(i,k), k] * REV[j] * T[n, j],  j = idx[blade(i) ^ blade(k)]
__device__ __forceinline__ float m_entry(int i, int k, const float* __restrict__ tn) {
    unsigned bi = nib(BLADE_PACK, (unsigned)i);
    unsigned bk = nib(BLADE_PACK, (unsigned)k);
    unsigned bj = bi ^ bk;
    unsigned j  = nib(IDX_PACK, bj);
    // Dorst reordering sign of blade(j)*blade(k), bj <= 0xF so 3 shifts suffice.
    int s = __popc((bj >> 1) & bk) + __popc((bj >> 2) & bk) + __popc((bj >> 3) & bk);
    // reverse() sign on j: (-1)^(g(g-1)/2), g = grade(bj)
    int g = __popc(bj);
    s += (g * (g - 1)) >> 1;
    float v = tn[j];                      // LDS gather
    v = (s & 1) ? -v : v;
    return (bj & bk & 1u) ? 0.0f : v;    // e0^2 = 0 annihilation
}

__global__ __launch_bounds__(256) void pga_cross_gp_kernel(
    const float* __restrict__ T,   // [B, N, 16]
    float* __restrict__ out,       // [B, N, N, 16]
    int N)
{
    __shared__ float tn[16];

    const int bn = blockIdx.x;            // b*N + n
    const int b  = bn / N;
    const int n  = bn - b * N;
    const float* Tb = T + (size_t)b * N * 16;

    if (threadIdx.x < 16) tn[threadIdx.x] = Tb[(size_t)n * 16 + threadIdx.x];
    __syncthreads();

    const int wave = threadIdx.x >> 5;    // 0..7
    const int lane = threadIdx.x & 31;
    const int li   = lane & 15;           // i (B cols / C cols) or m-row within half
    const int hi   = lane >> 4;           // half-wave select

    // Build the B operand (M_n^T in 4x16 K-chunks): bmat[kc].x -> K=4kc+2hi, .y -> K=4kc+2hi+1
    v2f bmat[4];
#pragma unroll
    for (int kc = 0; kc < 4; ++kc) {
        const int k0 = 4 * kc + 2 * hi;
        bmat[kc].x = m_entry(li, k0,     tn);
        bmat[kc].y = m_entry(li, k0 + 1, tn);
    }

    const size_t outbase = (size_t)bn * (size_t)N * 16;   // out[b, n, 0, 0]
    const int ntiles = N >> 4;                            // 128 m-tiles of 16 rows

    for (int mt = wave; mt < ntiles; mt += 8) {
        const int m0 = mt << 4;
        // A tile: rows m0+li, fp32 16x4 layout per chunk: VGPR0={K0,K2}, VGPR1={K1,K3}
        const float* arow = Tb + (size_t)(m0 + li) * 16 + 2 * hi;
        v2f a0 = *(const v2f*)(arow + 0);
        v2f a1 = *(const v2f*)(arow + 4);
        v2f a2 = *(const v2f*)(arow + 8);
        v2f a3 = *(const v2f*)(arow + 12);

        v8f c = {};
        c = __builtin_amdgcn_wmma_f32_16x16x4_f32(false, a0, false, bmat[0], (short)0, c, false, false);
        c = __builtin_amdgcn_wmma_f32_16x16x4_f32(false, a1, false, bmat[1], (short)0, c, false, false);
        c = __builtin_amdgcn_wmma_f32_16x16x4_f32(false, a2, false, bmat[2], (short)0, c, false, false);
        c = __builtin_amdgcn_wmma_f32_16x16x4_f32(false, a3, false, bmat[3], (short)0, c, false, false);

        // C layout: VGPR r = rows {m0+r (lanes 0-15), m0+r+8 (lanes 16-31)}, col = li.
        float* orow = out + outbase + ((size_t)(m0 + hi * 8) << 4) + li;
#pragma unroll
        for (int r = 0; r < 8; ++r) orow[(size_t)r << 4] = c[r];
    }
}

extern "C" void kernel_launch(void* const* d_in, const int* in_sizes, int n_in,
                              void* d_out, int out_size, void* d_ws, size_t ws_size,
                              hipStream_t stream) {
    (void)n_in; (void)out_size; (void)d_ws; (void)ws_size;
    const float* T = (const float*)d_in[0];
    float* out = (float*)d_out;
    const int B = 2;
    const int N = in_sizes[0] / (B * 16);   // 2048
    dim3 grid(B * N), block(256);
    pga_cross_gp_kernel<<<grid, block, 0, stream>>>(T, out, N);
}